// TopKRouter_44023414784362
// MI455X (gfx1250) — compile-verified
//
#include <hip/hip_runtime.h>
#include <hip/hip_bf16.h>
#include <stdint.h>

// ---------------- problem constants (match setup_inputs) ----------------
#define BATCH   32768
#define NM      64      // number of MLPs (M)
#define NH      128     // hidden (H)
#define HALF_N  (BATCH * NM / 2)   // 1048576: threefry iota split point
#define SUBT    4       // 16-row b-tiles per wave (weight reuse factor)

typedef float v2f __attribute__((ext_vector_type(2)));
typedef float v8f __attribute__((ext_vector_type(8)));

// ---------------- fast tanh (gfx1250 V_TANH_F32 when available) ----------
#if __has_builtin(__builtin_amdgcn_tanhf)
static __device__ __forceinline__ float fast_tanh(float x) { return __builtin_amdgcn_tanhf(x); }
#elif __has_builtin(__builtin_amdgcn_tanh_f32)
static __device__ __forceinline__ float fast_tanh(float x) { return __builtin_amdgcn_tanh_f32(x); }
#else
static __device__ __forceinline__ float fast_tanh(float x) { return tanhf(x); }
#endif

// =========================================================================
// Kernel 1: evaluate the MLP library -> mlp_out (B, 64) into workspace.
// One wave handles SUBT x 16 batch rows for one m via V_WMMA_F32_16X16X4_F32
// (K=2 real + 2 zero K-slots). Weight tiles are loaded once per ht and fed
// to SUBT WMMAs. All loads are unpredicated; lanes 16-31 are zeroed with
// selects (no EXEC branching in the hot loop).
// =========================================================================
__global__ __launch_bounds__(256) void mlp_eval_kernel(
    const float* __restrict__ X,   // (B, 2)
    const float* __restrict__ W1,  // (64, 2, 128)
    const float* __restrict__ b1,  // (64, 128)
    const float* __restrict__ W2,  // (64, 128)
    const float* __restrict__ b2,  // (64,)
    float* __restrict__ mlp_out)   // (B, 64)
{
  const int lane = threadIdx.x & 31;
  const int wave = threadIdx.x >> 5;
  const int b0   = blockIdx.x * (16 * SUBT);  // 64 rows of B per block
  const int m    = blockIdx.y * 8 + wave;     // one MLP per wave

  const int  col = lane & 15;                 // N (h sub-index); also A-row
  const bool lo  = lane < 16;

  // A tiles (16x4 fp32): lanes0-15 hold (K0,K1) for row=col; lanes16-31 -> (K2,K3)=0
  v2f a[SUBT];
#pragma unroll
  for (int s = 0; s < SUBT; ++s) {
    const v2f xv = ((const v2f*)X)[b0 + s * 16 + col];  // valid for all lanes
    a[s].x = lo ? xv.x : 0.0f;
    a[s].y = lo ? xv.y : 0.0f;
  }

  const float* w1m = W1 + (size_t)m * 2 * NH;
  const float* w2m = W2 + (size_t)m * NH;
  const float* b1m = b1 + (size_t)m * NH;

  float acc[SUBT][8];
#pragma unroll
  for (int s = 0; s < SUBT; ++s)
#pragma unroll
    for (int r = 0; r < 8; ++r) acc[s][r] = 0.0f;

#pragma unroll
  for (int ht = 0; ht < 8; ++ht) {            // H = 8 tiles of 16
    const int h0 = ht * 16;
    // B matrix 4x16 fp32: lanes0-15 carry rows K0,K1 (col=lane); lanes16-31 -> K2,K3=0
    const float w1k0 = w1m[h0 + col];
    const float w1k1 = w1m[NH + h0 + col];
    v2f bm;
    bm.x = lo ? w1k0 : 0.0f;
    bm.y = lo ? w1k1 : 0.0f;

    const float bias = b1m[h0 + col];         // column-wise bias (same every row)
    const float w2v  = w2m[h0 + col];

#pragma unroll
    for (int s = 0; s < SUBT; ++s) {
      v8f c;
#pragma unroll
      for (int r = 0; r < 8; ++r) c[r] = bias;
      // pre-activation tile: rows = b (8 VGPRs x 2 lane-halves), cols = h
      c = __builtin_amdgcn_wmma_f32_16x16x4_f32(false, a[s], false, bm,
                                                (short)0, c, false, false);
#pragma unroll
      for (int r = 0; r < 8; ++r) acc[s][r] += fast_tanh(c[r]) * w2v;
    }
  }

  // Reduce over the 16 column-lanes within each half-wave (sum over h), store.
  const float bb = b2[m];
#pragma unroll
  for (int s = 0; s < SUBT; ++s) {
#pragma unroll
    for (int r = 0; r < 8; ++r) {
      float v = acc[s][r];
      v += __shfl_xor(v, 1, 32);
      v += __shfl_xor(v, 2, 32);
      v += __shfl_xor(v, 4, 32);
      v += __shfl_xor(v, 8, 32);
      acc[s][r] = v;
    }
    if ((lane & 15) == 0) {
      const int rbase = b0 + s * 16 + (lane >> 4) * 8; // lane0: rows 0..7, lane16: 8..15
#pragma unroll
      for (int r = 0; r < 8; ++r)
        mlp_out[(size_t)(rbase + r) * NM + m] = acc[s][r] + bb;
    }
  }
}

// =========================================================================
// Threefry-2x32 (exact JAX schedule)
// =========================================================================
static __device__ __forceinline__ uint32_t rotl32(uint32_t x, int n) {
  return (x << n) | (x >> (32 - n));
}

static __device__ __forceinline__ void threefry2x32(uint32_t k0, uint32_t k1,
                                                    uint32_t& x0, uint32_t& x1) {
  const uint32_t ks2 = k0 ^ k1 ^ 0x1BD11BDAu;
  x0 += k0; x1 += k1;
#define TF_ROUND(r) { x0 += x1; x1 = rotl32(x1, (r)); x1 ^= x0; }
  TF_ROUND(13) TF_ROUND(15) TF_ROUND(26) TF_ROUND(6)
  x0 += k1;  x1 += ks2 + 1u;
  TF_ROUND(17) TF_ROUND(29) TF_ROUND(16) TF_ROUND(24)
  x0 += ks2; x1 += k0 + 2u;
  TF_ROUND(13) TF_ROUND(15) TF_ROUND(26) TF_ROUND(6)
  x0 += k0;  x1 += k1 + 3u;
  TF_ROUND(17) TF_ROUND(29) TF_ROUND(16) TF_ROUND(24)
  x0 += k1;  x1 += ks2 + 4u;
  TF_ROUND(13) TF_ROUND(15) TF_ROUND(26) TF_ROUND(6)
  x0 += ks2; x1 += k0 + 5u;
#undef TF_ROUND
}

static __device__ __forceinline__ float gumbel_from_bits(uint32_t bits) {
  // exact JAX uniform(minval=1e-10, maxval=1.0) in f32: (max-min) rounds to 1.0f
  float u = __uint_as_float((bits >> 9) | 0x3f800000u) - 1.0f;  // [0, 1)
  u = fmaxf(u + 1e-10f, 1e-10f);
  return -logf(-logf(u));
}

// =========================================================================
// Kernel 2: Gumbel top-2 routing. One wave per b_lo; the threefry call for
// count i = b_lo*64+m also yields bits for b_hi = b_lo + B/2 (iota halves).
// Hard gates forward-reduce to one-hot at argmax(logits + gumbel).
// =========================================================================
__global__ __launch_bounds__(256) void gate_router_kernel(
    const float* __restrict__ routers,  // (2, 64)
    const float* __restrict__ coeff,    // (2, 2, 64)
    const float* __restrict__ prior,    // (64,)
    const float* __restrict__ mlp_out,  // (B, 64)
    float* __restrict__ dxdt,           // (B, 2)
    float* __restrict__ gates)          // (2, 2, B, 64)
{
  const int lane = threadIdx.x & 31;
  const int wave = threadIdx.x >> 5;
  const int b_lo = blockIdx.x * 8 + wave;
  const int b_hi = b_lo + BATCH / 2;

  // folded keys: kj = threefry2x32(key(42)=(0,42), (0, d*2+j))
  uint32_t kk0[4], kk1[4];
#pragma unroll
  for (int t = 0; t < 4; ++t) {
    uint32_t x0 = 0u, x1 = (uint32_t)t;
    threefry2x32(0u, 42u, x0, x1);
    kk0[t] = x0; kk1[t] = x1;
  }

  const int m1 = lane, m2 = lane + 32;

#pragma unroll
  for (int d = 0; d < 2; ++d) {
    const float base1 = routers[d * NM + m1] + prior[m1];
    const float base2 = routers[d * NM + m2] + prior[m2];
    int sel_lo[2], sel_hi[2];

#pragma unroll
    for (int j = 0; j < 2; ++j) {
      const int t = d * 2 + j;
      uint32_t a0 = (uint32_t)(b_lo * NM + m1), a1 = a0 + HALF_N;
      threefry2x32(kk0[t], kk1[t], a0, a1);          // bits for (b_lo,m1),(b_hi,m1)
      uint32_t c0 = (uint32_t)(b_lo * NM + m2), c1 = c0 + HALF_N;
      threefry2x32(kk0[t], kk1[t], c0, c1);          // bits for (b_lo,m2),(b_hi,m2)

      float zl1 = base1 + gumbel_from_bits(a0);
      float zh1 = base1 + gumbel_from_bits(a1);
      float zl2 = base2 + gumbel_from_bits(c0);
      float zh2 = base2 + gumbel_from_bits(c1);

      if (j == 1) {   // -1e9 mask on the slot-0 winner
        if (m1 == sel_lo[0]) zl1 -= 1.0e9f;
        if (m2 == sel_lo[0]) zl2 -= 1.0e9f;
        if (m1 == sel_hi[0]) zh1 -= 1.0e9f;
        if (m2 == sel_hi[0]) zh2 -= 1.0e9f;
      }

      // lane-local best of the two m's (ties -> smaller index, m1 < m2)
      float v  = zl1; int idx  = m1; if (zl2 > v)  { v  = zl2; idx  = m2; }
      float vh = zh1; int idxh = m1; if (zh2 > vh) { vh = zh2; idxh = m2; }

      // wave-wide argmax, first-index tiebreak
#pragma unroll
      for (int off = 16; off >= 1; off >>= 1) {
        float ov  = __shfl_xor(v,   off, 32);
        int   oi  = __shfl_xor(idx, off, 32);
        if (ov > v || (ov == v && oi < idx)) { v = ov; idx = oi; }
        float ovh = __shfl_xor(vh,   off, 32);
        int   oih = __shfl_xor(idxh, off, 32);
        if (ovh > vh || (ovh == vh && oih < idxh)) { vh = ovh; idxh = oih; }
      }
      sel_lo[j] = idx; sel_hi[j] = idxh;

      // write full one-hot gate rows (coalesced across lanes)
      const size_t gb_lo = ((size_t)t * BATCH + (size_t)b_lo) * NM;
      const size_t gb_hi = ((size_t)t * BATCH + (size_t)b_hi) * NM;
      gates[gb_lo + m1] = (m1 == idx)  ? 1.0f : 0.0f;
      gates[gb_lo + m2] = (m2 == idx)  ? 1.0f : 0.0f;
      gates[gb_hi + m1] = (m1 == idxh) ? 1.0f : 0.0f;
      gates[gb_hi + m2] = (m2 == idxh) ? 1.0f : 0.0f;
    }

    if (lane < 2) {
      const int b  = lane ? b_hi : b_lo;
      const int s0 = lane ? sel_hi[0] : sel_lo[0];
      const int s1 = lane ? sel_hi[1] : sel_lo[1];
      const float r = coeff[(d * 2 + 0) * NM + s0] * mlp_out[(size_t)b * NM + s0]
                    + coeff[(d * 2 + 1) * NM + s1] * mlp_out[(size_t)b * NM + s1];
      dxdt[(size_t)b * 2 + d] = r;
    }
  }
}

// =========================================================================
extern "C" void kernel_launch(void* const* d_in, const int* in_sizes, int n_in,
                              void* d_out, int out_size, void* d_ws, size_t ws_size,
                              hipStream_t stream) {
  (void)in_sizes; (void)n_in; (void)out_size; (void)ws_size;

  const float* X       = (const float*)d_in[0];  // (B, 2)
  const float* W1      = (const float*)d_in[1];  // (64, 2, 128)
  const float* b1      = (const float*)d_in[2];  // (64, 128)
  const float* W2      = (const float*)d_in[3];  // (64, 128)
  const float* b2      = (const float*)d_in[4];  // (64,)
  const float* routers = (const float*)d_in[5];  // (2, 64)
  const float* coeff   = (const float*)d_in[6];  // (2, 2, 64)
  const float* prior   = (const float*)d_in[7];  // (64,)

  float* out    = (float*)d_out;
  float* dxdt   = out;                      // (B, 2) = 65536 floats
  float* gates  = out + (size_t)BATCH * 2;  // (2, 2, B, 64)
  float* mlp_ws = (float*)d_ws;             // (B, 64) scratch, 8 MB

  // Phase 1: MLP library via WMMA f32 16x16x4 (grid: 512 b-supertiles x 8 m-groups)
  dim3 g1(BATCH / (16 * SUBT), NM / 8);
  mlp_eval_kernel<<<g1, 256, 0, stream>>>(X, W1, b1, W2, b2, mlp_ws);

  // Phase 2: threefry Gumbel top-2 routing (one wave per b_lo; 8 waves/block)
  gate_router_kernel<<<(BATCH / 2) / 8, 256, 0, stream>>>(
      routers, coeff, prior, mlp_ws, dxdt, gates);
}